// DA_conv_26611617366320
// MI455X (gfx1250) — compile-verified
//
#include <hip/hip_runtime.h>
#include <math.h>

typedef __attribute__((ext_vector_type(2))) float v2f;
typedef __attribute__((ext_vector_type(8))) float v8f;

// ---------------------------------------------------------------------------
// Kernel 1: tiny MLPs from altitude.
//   feat[b][n] = lrelu(sum_k alt[b][k]*W1[n][k])          (8x128)
//   kern[b][m] = sum_k feat[b][k]*W2[m][k]                (8x1152) -> ws
//   a1[b][r]   = lrelu(sum_k alt[b][k]*ca1[r][k])         (8x16)
//   att[b][c]  = sigmoid(sum_r a1[b][r]*ca2[c][r])        (8x128)  -> ws
// ---------------------------------------------------------------------------
__global__ __launch_bounds__(256) void gen_kernel(
    const float* __restrict__ alt, const float* __restrict__ W1,
    const float* __restrict__ W2, const float* __restrict__ ca1,
    const float* __restrict__ ca2, float* __restrict__ kern_ws,
    float* __restrict__ att_ws)
{
  __shared__ float feat[8 * 128];
  __shared__ float a1s[8 * 16];
  const int tid = threadIdx.x;

  #pragma unroll
  for (int r = 0; r < 4; ++r) {
    int j = r * 256 + tid;                 // 1024 outputs
    int bb = j >> 7, n = j & 127;
    float s = 0.f;
    for (int k = 0; k < 128; ++k) s += alt[bb * 128 + k] * W1[n * 128 + k];
    feat[j] = (s > 0.f) ? s : 0.1f * s;
  }
  if (tid < 128) {                         // 8*16 outputs
    int bb = tid >> 4, r = tid & 15;
    float s = 0.f;
    for (int k = 0; k < 128; ++k) s += alt[bb * 128 + k] * ca1[r * 128 + k];
    a1s[tid] = (s > 0.f) ? s : 0.1f * s;
  }
  __syncthreads();

  for (int r = 0; r < 36; ++r) {
    int j = r * 256 + tid;                 // 9216 outputs
    int bb = j / 1152, m = j - bb * 1152;
    float s = 0.f;
    for (int k = 0; k < 128; ++k) s += feat[bb * 128 + k] * W2[m * 128 + k];
    kern_ws[j] = s;                        // layout [b][c][3][3] flat
  }
  #pragma unroll
  for (int r = 0; r < 4; ++r) {
    int j = r * 256 + tid;                 // 1024 outputs
    int bb = j >> 7, c = j & 127;
    float s = 0.f;
    #pragma unroll
    for (int k = 0; k < 16; ++k) s += a1s[bb * 16 + k] * ca2[c * 16 + k];
    att_ws[j] = 1.f / (1.f + expf(-s));
  }
}

// ---------------------------------------------------------------------------
// Kernel 2: fused depthwise(gen 3x3) + ReLU + 1x1 conv (WMMA f32) + bias
//           + residual x*att.  One block = (b, row h, 64-px half row).
// ---------------------------------------------------------------------------
__global__ __launch_bounds__(256) void fused_kernel(
    const float* __restrict__ x, const float* __restrict__ conv_w,
    const float* __restrict__ conv_b, const float* __restrict__ kern_ws,
    const float* __restrict__ att_ws, float* __restrict__ out)
{
  __shared__ float t[128 * 72];   // depthwise output tile [c][wl], stride 72
  __shared__ float kk[128 * 9];   // per-channel 3x3 kernels for this b
  __shared__ float att_s[128];
  __shared__ float cb_s[128];

  const int tid = threadIdx.x;
  const int b = blockIdx.x >> 8;
  const int rem = blockIdx.x & 255;
  const int h = rem >> 1;
  const int wbase = (rem & 1) * 64;

  for (int i = tid; i < 1152; i += 256) kk[i] = kern_ws[b * 1152 + i];
  if (tid < 128) {
    att_s[tid] = att_ws[b * 128 + tid];
    cb_s[tid] = conv_b[tid];
  }
  __syncthreads();

  // ---- Stage 1: generated depthwise 3x3 + ReLU -> LDS ----
  const float* xb = x + (size_t)b * (128 * 128 * 128);
  for (int it = 0; it < 32; ++it) {
    int idx = it * 256 + tid;              // 8192 = 128c x 64w
    int c = idx >> 6, wl = idx & 63;
    int w = wbase + wl;
    const float* xc = xb + (size_t)c * (128 * 128);
    const float* kc = kk + c * 9;          // uniform per wave -> broadcast
    float s = 0.f;
    #pragma unroll
    for (int dy = 0; dy < 3; ++dy) {
      int hh = h + dy - 1;
      if (hh >= 0 && hh < 128) {
        const float* xr = xc + hh * 128;
        #pragma unroll
        for (int dx = 0; dx < 3; ++dx) {
          int ww = w + dx - 1;
          if (ww >= 0 && ww < 128) s += kc[dy * 3 + dx] * xr[ww];
        }
      }
    }
    t[c * 72 + wl] = fmaxf(s, 0.f);
  }
  __syncthreads();

  // ---- Stage 2: 1x1 conv as GEMM via V_WMMA_F32_16X16X4_F32 ----
  // M = out channels (16 per wave), N = pixels (16 per tile), K = in channels.
  const int lane = tid & 31;
  const int half = lane >> 4;              // half-wave select
  const int lm = lane & 15;
  const int o0 = (tid >> 5) * 16;          // wave's output-channel tile

  // A fragments (conv_w[o][c]) per ISA 16x4 f32 layout:
  //   VGPR0: K = 2*half, VGPR1: K = 2*half+1, M = lane&15.
  v2f afrag[32];
  #pragma unroll
  for (int k4 = 0; k4 < 32; ++k4) {
    const int kb = k4 * 4 + 2 * half;
    const float* p = conv_w + (o0 + lm) * 128 + kb;
    afrag[k4].x = p[0];
    afrag[k4].y = p[1];
  }

  #pragma unroll
  for (int nt = 0; nt < 4; ++nt) {
    const int w0 = nt * 16;
    v8f acc = {};
    #pragma unroll
    for (int k4 = 0; k4 < 32; ++k4) {
      const int kb = k4 * 4 + 2 * half;
      v2f bfrag;                           // B[k][n] = t[kb..][w0+lm]
      bfrag.x = t[kb * 72 + w0 + lm];
      bfrag.y = t[(kb + 1) * 72 + w0 + lm];
      acc = __builtin_amdgcn_wmma_f32_16x16x4_f32(
          false, afrag[k4], false, bfrag, (short)0, acc, false, false);
    }
    // C/D layout: acc[v] holds M = v + 8*half, N = lane&15.
    const int w = wbase + w0 + lm;
    #pragma unroll
    for (int v = 0; v < 8; ++v) {
      const int o = o0 + v + half * 8;
      const size_t g = (((size_t)b * 128 + o) * 128 + h) * 128 + w;
      out[g] = acc[v] + cb_s[o] + x[g] * att_s[o];
    }
  }
}

extern "C" void kernel_launch(void* const* d_in, const int* in_sizes, int n_in,
                              void* d_out, int out_size, void* d_ws, size_t ws_size,
                              hipStream_t stream) {
  const float* x     = (const float*)d_in[0];
  const float* alt   = (const float*)d_in[1];
  const float* W1    = (const float*)d_in[2];
  const float* W2    = (const float*)d_in[3];
  const float* convw = (const float*)d_in[4];
  const float* convb = (const float*)d_in[5];
  const float* ca1   = (const float*)d_in[6];
  const float* ca2   = (const float*)d_in[7];
  float* outp = (float*)d_out;

  float* kern_ws = (float*)d_ws;            // 9216 floats
  float* att_ws  = kern_ws + 9216;          // 1024 floats

  gen_kernel<<<1, 256, 0, stream>>>(alt, W1, W2, ca1, ca2, kern_ws, att_ws);
  fused_kernel<<<2048, 256, 0, stream>>>(x, convw, convb, kern_ws, att_ws, outp);
}